// QCNN_Pure_29978871726313
// MI455X (gfx1250) — compile-verified
//
#include <hip/hip_runtime.h>
#include <math.h>

typedef float v2f __attribute__((ext_vector_type(2)));
typedef float v8f __attribute__((ext_vector_type(8)));

#define WAVES_PER_BLOCK 8
#define BLOCK_THREADS (WAVES_PER_BLOCK * 32)

// param block in d_ws:
//   [0..21]   : cos/sin pairs for crx[0..10]   (pp[2i], pp[2i+1])
//   [22..77]  : 7 U3 matrices, 8 floats each: U00.re,U00.im,U01.re,U01.im,U10.re,U10.im,U11.re,U11.im
#define PP_CRX   0
#define PP_U3    22
#define PP_TOTAL 78

struct cplx { float re, im; };
__device__ __forceinline__ cplx cmul(cplx a, cplx b) {
  return { a.re * b.re - a.im * b.im, a.re * b.im + a.im * b.re };
}
__device__ __forceinline__ cplx cadd(cplx a, cplx b) { return { a.re + b.re, a.im + b.im }; }

// ---- WMMA f32 16x16x4: D = A(16x4) x B(4x16) + C ----
__device__ __forceinline__ v8f wmma4(v2f a, v2f b, v8f c) {
  return __builtin_amdgcn_wmma_f32_16x16x4_f32(false, a, false, b, (short)0, c, false, false);
}

// Complex 16x16x16 matmul. A in A-layout (ar/ai/ani = -ai), B in B-layout (br/bi).
// Dre = Ar*Br - Ai*Bi ; Dim = Ar*Bi + Ai*Br. 16 v_wmma instructions.
__device__ __forceinline__ void cmatmul(const float* ar, const float* ai, const float* ani,
                                        const float* br, const float* bi,
                                        v8f& dre, v8f& dim) {
  v8f cr = {0.f,0.f,0.f,0.f,0.f,0.f,0.f,0.f};
  v8f ci = {0.f,0.f,0.f,0.f,0.f,0.f,0.f,0.f};
#pragma unroll
  for (int s = 0; s < 4; ++s) {
    v2f a_r  = { ar [2*s+0], ar [2*s+1] };
    v2f a_i  = { ai [2*s+0], ai [2*s+1] };
    v2f a_ni = { ani[2*s+0], ani[2*s+1] };
    v2f b_r  = { br [2*s+0], br [2*s+1] };
    v2f b_i  = { bi [2*s+0], bi [2*s+1] };
    cr = wmma4(a_r,  b_r, cr);
    cr = wmma4(a_ni, b_i, cr);
    ci = wmma4(a_r,  b_i, ci);
    ci = wmma4(a_i,  b_r, ci);
  }
  dre = cr; dim = ci;
}

// State (C/D layout) -> B-operand layout: slot j=2s+t holds S[4s+t+2*hi, n].
__device__ __forceinline__ void dlayout_to_b(const v8f& d, float* b, int hi) {
#pragma unroll
  for (int s = 0; s < 4; ++s) {
#pragma unroll
    for (int t = 0; t < 2; ++t) {
      int j = 2*s + t;
      float v;
      if (s < 2) {
        float other = __shfl_xor(d[4*s+t+2], 16, 32);
        v = hi ? other : d[4*s+t];
      } else {
        float other = __shfl_xor(d[4*s+t-8], 16, 32);
        v = hi ? d[4*s+t-6] : other;
      }
      b[j] = v;
    }
  }
}

// State (D layout) -> A-operand layout (full 16x16 transpose) via per-wave LDS region.
__device__ __forceinline__ void dlayout_to_a(const v8f& dre, const v8f& dim,
                                             float* ar, float* ai,
                                             float* lds, int n, int hi) {
#pragma unroll
  for (int v = 0; v < 8; ++v) {
    int row = v + 8*hi;
    lds[row*16 + n]       = dre[v];
    lds[256 + row*16 + n] = dim[v];
  }
  __asm__ volatile("s_wait_dscnt 0" ::: "memory");
#pragma unroll
  for (int j = 0; j < 8; ++j) {
    int k = 4*(j>>1) + (j&1) + 2*hi;
    ar[j] = lds[n*16 + k];
    ai[j] = lds[256 + n*16 + k];
  }
}

// ---- sparse gates, row side (register space: row = v + 8*hi) ----
__device__ __forceinline__ void crx_left(v8f& re, v8f& im, int cm, int rm, float c, float s) {
  v8f nre = re, nim = im;
#pragma unroll
  for (int v = 0; v < 8; ++v) {
    if (v & cm) {
      int p = v ^ rm;
      nre[v] = c*re[v] + s*im[p];
      nim[v] = c*im[v] - s*re[p];
    }
  }
  re = nre; im = nim;
}
__device__ __forceinline__ void crx_left_q0(v8f& re, v8f& im, int rm, float c, float s, int hi) {
  v8f nre, nim;
#pragma unroll
  for (int v = 0; v < 8; ++v) {
    int p = v ^ rm;
    float nr = c*re[v] + s*im[p];
    float ni = c*im[v] - s*re[p];
    nre[v] = hi ? nr : re[v];
    nim[v] = hi ? ni : im[v];
  }
  re = nre; im = nim;
}
__device__ __forceinline__ void u1_left(v8f& re, v8f& im, int rm, const cplx U[2][2]) {
  v8f nre, nim;
#pragma unroll
  for (int v = 0; v < 8; ++v) {
    int bsel = (v & rm) ? 1 : 0;
    int p = v ^ rm;
    cplx self = { re[v], im[v] };
    cplx part = { re[p], im[p] };
    cplx cs = bsel ? U[1][1] : U[0][0];
    cplx cp = bsel ? U[1][0] : U[0][1];
    cplx o = cadd(cmul(cs, self), cmul(cp, part));
    nre[v] = o.re; nim[v] = o.im;
  }
  re = nre; im = nim;
}

// ---- sparse gates, column side (lane space: col = lane&15) ----
__device__ __forceinline__ void crx_right(v8f& re, v8f& im, int cm, int tm,
                                          float c, float s, int n) {
  bool ctl = (n & cm) != 0;
#pragma unroll
  for (int v = 0; v < 8; ++v) {
    float pr = __shfl_xor(re[v], tm, 32);
    float pi = __shfl_xor(im[v], tm, 32);
    float nr = c*re[v] + s*pi;
    float ni = c*im[v] - s*pr;
    re[v] = ctl ? nr : re[v];
    im[v] = ctl ? ni : im[v];
  }
}
__device__ __forceinline__ void crx_cross(v8f& re, v8f& im, int tm, float c, float s) {
#pragma unroll
  for (int v = 1; v < 8; v += 2) {
    float pr = __shfl_xor(re[v], tm, 32);
    float pi = __shfl_xor(im[v], tm, 32);
    re[v] = c*re[v] + s*pi;
    im[v] = c*im[v] - s*pr;
  }
}
__device__ __forceinline__ void u1_right(v8f& re, v8f& im, int tm, const cplx U[2][2], int n) {
  int bsel = (n & tm) ? 1 : 0;
  cplx cs = bsel ? U[1][1] : U[0][0];
  cplx cp = bsel ? U[1][0] : U[0][1];
#pragma unroll
  for (int v = 0; v < 8; ++v) {
    float pr = __shfl_xor(re[v], tm, 32);
    float pi = __shfl_xor(im[v], tm, 32);
    cplx self = { re[v], im[v] };
    cplx part = { pr, pi };
    cplx o = cadd(cmul(cs, self), cmul(cp, part));
    re[v] = o.re; im[v] = o.im;
  }
}

__device__ __forceinline__ void load_u3(const float* __restrict__ p, cplx U[2][2]) {
  U[0][0] = { p[0], p[1] };
  U[0][1] = { p[2], p[3] };
  U[1][0] = { p[4], p[5] };
  U[1][1] = { p[6], p[7] };
}

// ===== setup kernel: batch-invariant trig, computed once (one warp) =====
__global__ void qcnn_setup(const float* __restrict__ crx, const float* __restrict__ u3p,
                           float* __restrict__ pp) {
  int t = threadIdx.x;
  if (t < 11) {
    float a = 0.5f * crx[t];
    pp[PP_CRX + 2*t + 0] = __cosf(a);
    pp[PP_CRX + 2*t + 1] = __sinf(a);
  } else if (t < 18) {
    int i = t - 11;
    float th = u3p[3*i + 0], ph = u3p[3*i + 1], lm = u3p[3*i + 2];
    float ct = __cosf(0.5f*th), st = __sinf(0.5f*th);
    float* q = pp + PP_U3 + 8*i;
    q[0] =  ct;              q[1] = 0.f;                // U00
    q[2] = -st*__cosf(lm);   q[3] = -st*__sinf(lm);     // U01
    q[4] =  st*__cosf(ph);   q[5] =  st*__sinf(ph);     // U10
    q[6] =  ct*__cosf(ph+lm);q[7] =  ct*__sinf(ph+lm);  // U11
  }
}

__global__ __launch_bounds__(BLOCK_THREADS)
void qcnn_kernel(const float* __restrict__ theta, const float* __restrict__ phi,
                 const float* __restrict__ pp, float* __restrict__ out, int B) {
  __shared__ float lds_all[WAVES_PER_BLOCK * 512];
  const int tid  = threadIdx.x;
  const int wave = tid >> 5;
  const int l    = tid & 31;
  const int hi   = l >> 4;
  const int n    = l & 15;
  float* lds = &lds_all[wave * 512];
  const int b = blockIdx.x * WAVES_PER_BLOCK + wave;
  if (b >= B) return;   // wave-uniform

  // --- per-batch 2x2: V = RZ(phi) * RY(theta) ---
  float th = theta[b], ph = phi[b];
  float ct = __cosf(0.5f*th), st = __sinf(0.5f*th);
  float cp = __cosf(0.5f*ph), sp = __sinf(0.5f*ph);
  cplx V[2][2];
  V[0][0] = {  ct*cp, -ct*sp };
  V[0][1] = { -st*cp,  st*sp };
  V[1][0] = {  st*cp,  st*sp };
  V[1][1] = {  ct*cp,  ct*sp };

  // Left cycle operator L~ = P_cnot(0,1)(1,2)(2,3) * V^(x4), A-layout.
  float la_r[8], la_i[8], la_ni[8];
#pragma unroll
  for (int j = 0; j < 8; ++j) {
    int k = 4*(j>>1) + (j&1) + 2*hi;
    int m3=(n>>3)&1, m2=(n>>2)&1, m1=(n>>1)&1, m0=n&1;
    int r3=m3, r2=m2^m3, r1=m1^m2, r0=m0^m1;      // P^-1(row)
    int k3=(k>>3)&1, k2=(k>>2)&1, k1=(k>>1)&1, k0=k&1;
    cplx e = cmul(cmul(V[r3][k3], V[r2][k2]), cmul(V[r1][k1], V[r0][k0]));
    la_r[j] = e.re; la_i[j] = e.im; la_ni[j] = -e.im;
  }
  // Right cycle operator (V^(x4))^T in B-layout: slot holds (V^x4)[n, k].
  float rb_r[8], rb_i[8];
#pragma unroll
  for (int j = 0; j < 8; ++j) {
    int k = 4*(j>>1) + (j&1) + 2*hi;
    int c3=(n>>3)&1, c2=(n>>2)&1, c1=(n>>1)&1, c0=n&1;
    int k3=(k>>3)&1, k2=(k>>2)&1, k1=(k>>1)&1, k0=k&1;
    cplx e = cmul(cmul(V[c3][k3], V[c2][k2]), cmul(V[c1][k1], V[c0][k0]));
    rb_r[j] = e.re; rb_i[j] = e.im;
  }

  // Column CNOT-chain (4,5)(5,6)(6,7) permutation: dest lane's source lane.
  int qa3=(n>>3)&1, qa2=(n>>2)&1, qa1=(n>>1)&1, qa0=n&1;
  int qsrc  = (qa3<<3) | ((qa2^qa3)<<2) | ((qa1^qa2)<<1) | (qa0^qa1);
  int qsrcl = (hi<<4) | qsrc;

  // |0...0>
  v8f s_re = {0.f,0.f,0.f,0.f,0.f,0.f,0.f,0.f};
  v8f s_im = {0.f,0.f,0.f,0.f,0.f,0.f,0.f,0.f};
  s_re[0] = (l == 0) ? 1.f : 0.f;

  // ================= 4 entangling cycles =================
  for (int cyc = 0; cyc < 4; ++cyc) {
    float sbr[8], sbi[8];
    dlayout_to_b(s_re, sbr, hi);
    dlayout_to_b(s_im, sbi, hi);
    v8f t_re, t_im;
    cmatmul(la_r, la_i, la_ni, sbr, sbi, t_re, t_im);

    float tar[8], tai[8], tani[8];
    dlayout_to_a(t_re, t_im, tar, tai, lds, n, hi);
#pragma unroll
    for (int j = 0; j < 8; ++j) tani[j] = -tai[j];
    cmatmul(tar, tai, tani, rb_r, rb_i, s_re, s_im);

    // CNOT(3,4): odd rows swap columns n <-> n^8
#pragma unroll
    for (int v = 1; v < 8; v += 2) {
      s_re[v] = __shfl_xor(s_re[v], 8, 32);
      s_im[v] = __shfl_xor(s_im[v], 8, 32);
    }
    // CNOT(4,5)(5,6)(6,7): static column permutation
#pragma unroll
    for (int v = 0; v < 8; ++v) {
      s_re[v] = __shfl(s_re[v], qsrcl, 32);
      s_im[v] = __shfl(s_im[v], qsrcl, 32);
    }
    // CNOT(7,0): odd columns swap rows m <-> m^8 (lane-half swap)
#pragma unroll
    for (int v = 0; v < 8; ++v) {
      float tr = __shfl_xor(s_re[v], 16, 32);
      float ti = __shfl_xor(s_im[v], 16, 32);
      s_re[v] = (n & 1) ? tr : s_re[v];
      s_im[v] = (n & 1) ? ti : s_im[v];
    }
  }

  // ================= QCNN layer 1 =================
#define CRX_C(i) pp[PP_CRX + 2*(i) + 0]
#define CRX_S(i) pp[PP_CRX + 2*(i) + 1]
  crx_left_q0(s_re, s_im, 4, CRX_C(0), CRX_S(0), hi);   // CRX(0,1)
  crx_left   (s_re, s_im, 2, 1, CRX_C(1), CRX_S(1));    // CRX(2,3)
  crx_right  (s_re, s_im, 8, 4, CRX_C(2), CRX_S(2), n); // CRX(4,5)
  crx_right  (s_re, s_im, 2, 1, CRX_C(3), CRX_S(3), n); // CRX(6,7)
  crx_left   (s_re, s_im, 4, 2, CRX_C(4), CRX_S(4));    // CRX(1,2)
  crx_right  (s_re, s_im, 4, 2, CRX_C(6), CRX_S(6), n); // CRX(5,6)
  crx_cross  (s_re, s_im, 8, CRX_C(5), CRX_S(5));       // CRX(3,4) (commutes past later ones)

  cplx U0[2][2], U1[2][2], U2[2][2], U3m[2][2];
  load_u3(pp + PP_U3 + 8*0, U0);   // wire 1
  load_u3(pp + PP_U3 + 8*1, U1);   // wire 3
  load_u3(pp + PP_U3 + 8*2, U2);   // wire 5
  load_u3(pp + PP_U3 + 8*3, U3m);  // wire 7
  u1_left (s_re, s_im, 4, U0);
  u1_left (s_re, s_im, 1, U1);
  u1_right(s_re, s_im, 4, U2,  n);
  u1_right(s_re, s_im, 1, U3m, n);

  // ================= QCNN layer 2 =================
  crx_left (s_re, s_im, 4, 1, CRX_C(7), CRX_S(7));      // CRX(1,3)
  crx_right(s_re, s_im, 4, 1, CRX_C(8), CRX_S(8), n);   // CRX(5,7)
  crx_cross(s_re, s_im, 4, CRX_C(9), CRX_S(9));         // CRX(3,5)

  cplx U4[2][2], U5[2][2];
  load_u3(pp + PP_U3 + 8*4, U4);   // wire 3
  load_u3(pp + PP_U3 + 8*5, U5);   // wire 7
  u1_left (s_re, s_im, 1, U4);
  u1_right(s_re, s_im, 1, U5, n);

  // ================= QCNN layer 3 =================
  crx_cross(s_re, s_im, 1, CRX_C(10), CRX_S(10));       // CRX(3,7)
  cplx U6[2][2];
  load_u3(pp + PP_U3 + 8*6, U6);   // wire 7
  u1_right(s_re, s_im, 1, U6, n);
#undef CRX_C
#undef CRX_S

  // ================= <Z_7>, relu, store =================
  float acc = 0.f;
#pragma unroll
  for (int v = 0; v < 8; ++v) acc += s_re[v]*s_re[v] + s_im[v]*s_im[v];
  float pz = (n & 1) ? -acc : acc;          // q7 = column LSB
#pragma unroll
  for (int off = 1; off < 32; off <<= 1) pz += __shfl_xor(pz, off, 32);
  if (l == 0) out[b] = fmaxf(pz, 0.f);
}

extern "C" void kernel_launch(void* const* d_in, const int* in_sizes, int n_in,
                              void* d_out, int out_size, void* d_ws, size_t ws_size,
                              hipStream_t stream) {
  const float* theta = (const float*)d_in[0];
  const float* phi   = (const float*)d_in[1];
  const float* crx   = (const float*)d_in[2];
  const float* u3p   = (const float*)d_in[3];
  float* out = (float*)d_out;
  float* pp  = (float*)d_ws;
  const int B = in_sizes[0];

  qcnn_setup<<<1, 32, 0, stream>>>(crx, u3p, pp);

  const int blocks = (B + WAVES_PER_BLOCK - 1) / WAVES_PER_BLOCK;
  qcnn_kernel<<<blocks, BLOCK_THREADS, 0, stream>>>(theta, phi, pp, out, B);
}